// ScaledDotProductAttention_54752243090126
// MI455X (gfx1250) — compile-verified
//
#include <hip/hip_runtime.h>
#include <hip/hip_bf16.h>
#include <math.h>

typedef __attribute__((ext_vector_type(2))) float v2f;
typedef __attribute__((ext_vector_type(8))) float v8f;

#define B_SZ 8
#define TQ 1024
#define TK 2048
#define D_MODEL 512
#define TWO_D 1024

// Async copy of 16 bytes/lane from global to LDS (ASYNCcnt-tracked, CDNA5).
__device__ __forceinline__ void async_b128(unsigned lds_off, const float* gptr) {
  asm volatile("global_load_async_to_lds_b128 %0, %1, off"
               :: "v"(lds_off), "v"(gptr) : "memory");
}
__device__ __forceinline__ void wait_async0() {
  asm volatile("s_wait_asynccnt 0x0" ::: "memory");
}
__device__ __forceinline__ unsigned lds_addr(const void* p) {
  // generic (flat) pointer to a __shared__ object: low 32 bits = LDS byte offset
  return (unsigned)(unsigned long long)p;
}

// ---------------------------------------------------------------------------
// fp32 WMMA GEMM with LDS double buffering + async global->LDS copies.
//   C = A * op(B) (+bias[col]) (+log(mask[col]))
//   A:  [M x K] row-major.
//   TRANS_B=1: B is [N x K] row-major  -> C = A * B^T   (NT)
//   TRANS_B=0: B is [K x N] row-major  -> C = A * B     (NN)
// blockDim (32,4): 4 waves; block computes 64x64 of C; each wave 16x64.
// K tiled by 32. M,N multiples of 64; K multiple of 32 (true for all calls).
// ---------------------------------------------------------------------------
template <bool TRANS_B, bool ADD_BIAS, bool ADD_MASKLOG>
__global__ __launch_bounds__(128)
void wmma_gemm_f32(const float* __restrict__ A, int lda, long long strideA,
                   const float* __restrict__ Bm, int ldb, long long strideB,
                   float* __restrict__ C, int ldc, long long strideC,
                   const float* __restrict__ bias,
                   const float* __restrict__ mask, long long strideMask,
                   int M, int N, int K) {
  constexpr int AS = 36;                       // A tile LDS stride (floats), 16B aligned, bank-safe
  constexpr int BS = TRANS_B ? 36 : 68;        // B tile LDS stride
  constexpr int BROWS = TRANS_B ? 64 : 32;     // B tile rows in LDS

  __shared__ float Ab[2][64 * AS];
  __shared__ float Bb[2][BROWS * BS];

  const int lane  = threadIdx.x;               // 0..31
  const int wave  = threadIdx.y;               // 0..3
  const int tid   = wave * 32 + lane;          // 0..127
  const int batch = blockIdx.z;

  A  += (long long)batch * strideA;
  Bm += (long long)batch * strideB;
  C  += (long long)batch * strideC;

  const int m_blk = blockIdx.y * 64;
  const int n_blk = blockIdx.x * 64;

  // --- async tile loader: K-chunk kt -> LDS buffer buf -------------------
  auto load_tiles = [&](int kt, int buf) {
    const int k0 = kt * 32;
    // A tile: 64 rows x 32 floats = 512 x 16B chunks (8 chunks/row), 4/thread
#pragma unroll
    for (int i = 0; i < 4; ++i) {
      const int idx = tid + i * 128;
      const int row = idx >> 3;
      const int cc  = (idx & 7) * 4;
      async_b128(lds_addr(&Ab[buf][row * AS + cc]),
                 A + (long long)(m_blk + row) * lda + k0 + cc);
    }
    if (TRANS_B) {
      // B tile: [n][k] 64 rows x 32 floats
#pragma unroll
      for (int i = 0; i < 4; ++i) {
        const int idx = tid + i * 128;
        const int row = idx >> 3;
        const int cc  = (idx & 7) * 4;
        async_b128(lds_addr(&Bb[buf][row * BS + cc]),
                   Bm + (long long)(n_blk + row) * ldb + k0 + cc);
      }
    } else {
      // B tile: [k][n] 32 rows x 64 floats = 512 x 16B chunks (16/row)
#pragma unroll
      for (int i = 0; i < 4; ++i) {
        const int idx = tid + i * 128;
        const int row = idx >> 4;
        const int cc  = (idx & 15) * 4;
        async_b128(lds_addr(&Bb[buf][row * BS + cc]),
                   Bm + (long long)(k0 + row) * ldb + n_blk + cc);
      }
    }
  };

  const int half = lane >> 4;                  // f32 WMMA K-pair selector
  const int l16  = lane & 15;

  v8f acc0 = {}, acc1 = {}, acc2 = {}, acc3 = {};

  const int NKT = K / 32;
  load_tiles(0, 0);
  wait_async0();
  __syncthreads();

  for (int kt = 0; kt < NKT; ++kt) {
    const int cur = kt & 1;
    if (kt + 1 < NKT) load_tiles(kt + 1, cur ^ 1);   // prefetch next chunk

    const float* Ar = &Ab[cur][(wave * 16 + l16) * AS + 2 * half];
#pragma unroll
    for (int kk = 0; kk < 32; kk += 4) {
      v2f a = *(const v2f*)(Ar + kk);
      v2f b0, b1, b2, b3;
      if (TRANS_B) {
        const float* Bp = &Bb[cur][kk + 2 * half];
        b0 = *(const v2f*)(Bp + (l16 +  0) * BS);
        b1 = *(const v2f*)(Bp + (l16 + 16) * BS);
        b2 = *(const v2f*)(Bp + (l16 + 32) * BS);
        b3 = *(const v2f*)(Bp + (l16 + 48) * BS);
      } else {
        const float* Bp = &Bb[cur][(kk + 2 * half) * BS];
        b0.x = Bp[l16 +  0]; b0.y = Bp[BS + l16 +  0];
        b1.x = Bp[l16 + 16]; b1.y = Bp[BS + l16 + 16];
        b2.x = Bp[l16 + 32]; b2.y = Bp[BS + l16 + 32];
        b3.x = Bp[l16 + 48]; b3.y = Bp[BS + l16 + 48];
      }
      acc0 = __builtin_amdgcn_wmma_f32_16x16x4_f32(false, a, false, b0, (short)0, acc0, false, false);
      acc1 = __builtin_amdgcn_wmma_f32_16x16x4_f32(false, a, false, b1, (short)0, acc1, false, false);
      acc2 = __builtin_amdgcn_wmma_f32_16x16x4_f32(false, a, false, b2, (short)0, acc2, false, false);
      acc3 = __builtin_amdgcn_wmma_f32_16x16x4_f32(false, a, false, b3, (short)0, acc3, false, false);
    }

    wait_async0();        // own async copies into next buffer landed
    __syncthreads();      // everyone done reading cur / writing next
  }

  // --- epilogue: D row = m + v + 8*half, col = n_blk + 16*j + l16 --------
  const int m0 = m_blk + wave * 16;
  int   cols[4];
  float addv[4] = {0.f, 0.f, 0.f, 0.f};
#pragma unroll
  for (int j = 0; j < 4; ++j) cols[j] = n_blk + 16 * j + l16;
  if (ADD_BIAS) {
#pragma unroll
    for (int j = 0; j < 4; ++j) addv[j] += bias[cols[j]];
  }
  if (ADD_MASKLOG) {
    const float* mk = mask + (long long)batch * strideMask;
#pragma unroll
    for (int j = 0; j < 4; ++j) addv[j] += __logf(mk[cols[j]]);
  }
#pragma unroll
  for (int v = 0; v < 8; ++v) {
    float* crow = C + (long long)(m0 + v + 8 * half) * ldc;
    crow[cols[0]] = acc0[v] + addv[0];
    crow[cols[1]] = acc1[v] + addv[1];
    crow[cols[2]] = acc2[v] + addv[2];
    crow[cols[3]] = acc3[v] + addv[3];
  }
}

// ---------------------------------------------------------------------------
// In-place row softmax over 2048 columns. One 256-thread (8-wave) block / row.
// ---------------------------------------------------------------------------
__global__ __launch_bounds__(256)
void softmax2048_kernel(float* __restrict__ S) {
  __shared__ float red[8];
  float* p = S + (long long)blockIdx.x * 2048;
  const int t = threadIdx.x;

  float v[8];
#pragma unroll
  for (int i = 0; i < 8; ++i) v[i] = p[t + 256 * i];

  float m = v[0];
#pragma unroll
  for (int i = 1; i < 8; ++i) m = fmaxf(m, v[i]);
#pragma unroll
  for (int off = 16; off > 0; off >>= 1) m = fmaxf(m, __shfl_xor(m, off, 32));
  if ((t & 31) == 0) red[t >> 5] = m;
  __syncthreads();
  float mrow = red[0];
#pragma unroll
  for (int j = 1; j < 8; ++j) mrow = fmaxf(mrow, red[j]);
  __syncthreads();

  float s = 0.f;
#pragma unroll
  for (int i = 0; i < 8; ++i) { v[i] = __expf(v[i] - mrow); s += v[i]; }
#pragma unroll
  for (int off = 16; off > 0; off >>= 1) s += __shfl_xor(s, off, 32);
  if ((t & 31) == 0) red[t >> 5] = s;
  __syncthreads();
  float stot = 0.f;
#pragma unroll
  for (int j = 0; j < 8; ++j) stot += red[j];

  const float inv = 1.0f / stot;
#pragma unroll
  for (int i = 0; i < 8; ++i) p[t + 256 * i] = v[i] * inv;
}

// ---------------------------------------------------------------------------
extern "C" void kernel_launch(void* const* d_in, const int* in_sizes, int n_in,
                              void* d_out, int out_size, void* d_ws, size_t ws_size,
                              hipStream_t stream) {
  const float* dec     = (const float*)d_in[0];  // [8,1024,512]
  //                     d_in[1] decoder_mask (unused by reference)
  const float* estates = (const float*)d_in[2];  // [8,2048,1024]
  const float* emask   = (const float*)d_in[3];  // [8,2048]
  const float* W       = (const float*)d_in[4];  // [512,1024]
  const float* bproj   = (const float*)d_in[5];  // [512]

  float* ctx  = (float*)d_out;                                // [8,1024,512]
  float* attn = (float*)d_out + (size_t)B_SZ * TQ * D_MODEL;  // [8,1024,2048]
  float* enc  = (float*)d_ws;                                 // [8,2048,512] = 32 MB

  dim3 blk(32, 4);

  // 1) enc = E @ W^T + b        (M=16384, N=512, K=1024, NT)
  {
    dim3 grid(D_MODEL / 64, (B_SZ * TK) / 64, 1);
    wmma_gemm_f32<true, true, false><<<grid, blk, 0, stream>>>(
        estates, TWO_D, 0, W, TWO_D, 0, enc, D_MODEL, 0,
        bproj, nullptr, 0, B_SZ * TK, D_MODEL, TWO_D);
  }

  // 2) S = Q @ enc^T + log(mask)   per batch (M=1024, N=2048, K=512, NT)
  {
    dim3 grid(TK / 64, TQ / 64, B_SZ);
    wmma_gemm_f32<true, false, true><<<grid, blk, 0, stream>>>(
        dec, D_MODEL, (long long)TQ * D_MODEL,
        enc, D_MODEL, (long long)TK * D_MODEL,
        attn, TK, (long long)TQ * TK,
        nullptr, emask, TK, TQ, TK, D_MODEL);
  }

  // 3) softmax over k, in place on attn (8192 rows x 2048)
  softmax2048_kernel<<<B_SZ * TQ, 256, 0, stream>>>(attn);

  // 4) ctx = P @ enc               per batch (M=1024, N=512, K=2048, NN)
  {
    dim3 grid(D_MODEL / 64, TQ / 64, B_SZ);
    wmma_gemm_f32<false, false, false><<<grid, blk, 0, stream>>>(
        attn, TK, (long long)TQ * TK,
        enc, D_MODEL, (long long)TK * D_MODEL,
        ctx, D_MODEL, (long long)TQ * D_MODEL,
        nullptr, nullptr, 0, TQ, D_MODEL, TK);
  }
}